// StateEstimator_13580686590047
// MI455X (gfx1250) — compile-verified
//
#include <hip/hip_runtime.h>

typedef __attribute__((ext_vector_type(2))) float v2f;
typedef __attribute__((ext_vector_type(4))) float v4f;
typedef __attribute__((ext_vector_type(8))) float v8f;

#define WMMA4(a, b, c) \
  __builtin_amdgcn_wmma_f32_16x16x4_f32(false, (a), false, (b), (short)0, (c), false, false)

namespace {
constexpr int   kT   = 512;
constexpr int   kDin = 16;
constexpr int   kH   = 64;
constexpr int   kOut = 6;
constexpr float kAfb = 0.999f;
constexpr float kNfb = 1.0f - 0.999f;
}

// One wave (32 lanes) owns a 16-row batch tile and runs the full T=512
// recurrence with all weights and neuron state register-resident.
__global__ __launch_bounds__(32)
void snn_seq_kernel(const float* __restrict__ x,
                    const float* __restrict__ W1,
                    const float* __restrict__ b1,
                    const float* __restrict__ W2,
                    const float* __restrict__ b2,
                    const float* __restrict__ betas,
                    const float* __restrict__ thresholds,
                    const float* __restrict__ thr_int,
                    float* __restrict__ out)
{
  // spike transpose buffer: [k = 0..127][m = 0..15], k-major, 8 KB
  __shared__ float spk[2 * kH * 16];

  const int  lane = threadIdx.x;
  const int  n16  = lane & 15;
  const bool hi   = lane >= 16;
  const int  koff = hi ? 2 : 0;        // f32 WMMA A/B: lanes 16-31 hold K+2,K+3
  const int  m0   = blockIdx.x * 16;   // batch tile base

  // ---- constant B operand of GEMM1: W1^T as 4 N-tiles x 4 K-chunks ----
  // B 4x16 layout: lane n = N, vgpr0 = row K(+koff), vgpr1 = row K+1(+koff)
  v2f w1v[4][4];
  #pragma unroll
  for (int tt = 0; tt < 4; ++tt)
    #pragma unroll
    for (int c = 0; c < 4; ++c) {
      const float* w = W1 + (16 * tt + n16) * kDin + 4 * c + koff;
      v2f v; v.x = w[0]; v.y = w[1];
      w1v[tt][c] = v;
    }

  // ---- constant A operand of GEMM2: W2 zero-padded to 16 rows, 32 K-chunks ----
  // A 16x4 layout: lane%16 = M (output channel), x/y = K+koff, K+koff+1
  v2f w2v[32];
  #pragma unroll
  for (int c = 0; c < 32; ++c) {
    v2f v; v.x = 0.f; v.y = 0.f;
    if (n16 < kOut) {
      const float* w = W2 + n16 * (2 * kH) + 4 * c + koff;
      v.x = w[0]; v.y = w[1];
    }
    w2v[c] = v;
  }

  // ---- per-lane neuron params: H index = 16*tt + n16 (GEMM1 D: lane = N = H) ----
  float beta[4], thr[4], thri[4], hb[4];
  #pragma unroll
  for (int tt = 0; tt < 4; ++tt) {
    const int hI = 16 * tt + n16;
    beta[tt] = fminf(fmaxf(betas[hI], 0.f), 1.f);  // snntorch clamp
    thr[tt]  = thresholds[hI];
    thri[tt] = thr_int[hI];
    hb[tt]   = b1[hI];
    // alphas[] multiplies an all-zero syn state in the reference (dead code).
  }

  // ---- b2 bias in y^T D-layout: lanes 0-15 hold channels r=0..7, rest padding ----
  v8f ybias, zero8;
  #pragma unroll
  for (int r = 0; r < 8; ++r) {
    ybias[r] = (!hi && r < kOut) ? b2[r] : 0.f;
    zero8[r] = 0.f;
  }

  // ---- recurrent state ----
  v8f msyn[4], mint[4], yv;
  #pragma unroll
  for (int r = 0; r < 8; ++r) yv[r] = 0.f;
  #pragma unroll
  for (int tt = 0; tt < 4; ++tt)
    #pragma unroll
    for (int r = 0; r < 8; ++r) { msyn[tt][r] = 0.f; mint[tt][r] = 0.f; }

  // x row for this lane's batch element (lane halves need K+koff, K+koff+1 pairs)
  const float* xrow = x + (size_t)(m0 + n16) * kT * kDin + koff;
  v2f xp[4];
  #pragma unroll
  for (int c = 0; c < 4; ++c) xp[c] = *(const v2f*)(xrow + 4 * c);

  #pragma unroll 1
  for (int t = 0; t < kT; ++t) {
    // ---- output feedback mix into last 6 inputs (A-matrix chunks 2,3) ----
    // y^T D-layout: batch n's channels 0..7 live in lane n (low half).
    const float g0 = __shfl_xor(yv[0], 16, 32);
    const float g1 = __shfl_xor(yv[1], 16, 32);
    const float g4 = __shfl_xor(yv[4], 16, 32);
    const float g5 = __shfl_xor(yv[5], 16, 32);
    v2f a1[4];
    a1[0] = xp[0];
    a1[1] = xp[1];
    if (!hi) {                                   // K = 8,9 raw; K = 12,13 -> out[2],out[3]
      a1[2] = xp[2];
      v2f v; v.x = kAfb * yv[2] + kNfb * xp[3].x;
             v.y = kAfb * yv[3] + kNfb * xp[3].y;
      a1[3] = v;
    } else {                                     // K = 10,11 -> out[0],out[1]; K = 14,15 -> out[4],out[5]
      v2f u; u.x = kAfb * g0 + kNfb * xp[2].x;
             u.y = kAfb * g1 + kNfb * xp[2].y;
      a1[2] = u;
      v2f v; v.x = kAfb * g4 + kNfb * xp[3].x;
             v.y = kAfb * g5 + kNfb * xp[3].y;
      a1[3] = v;
    }

    // ---- software-pipeline next step's x; prefetch farther ahead ----
    if (t + 1 < kT) {
      const float* xn = xrow + (size_t)(t + 1) * kDin;
      #pragma unroll
      for (int c = 0; c < 4; ++c) xp[c] = *(const v2f*)(xn + 4 * c);
      if (t + 8 < kT) __builtin_prefetch(xrow + (size_t)(t + 8) * kDin, 0, 0);
    }

    // ---- GEMM1: h[16b x 64H] = x_mod @ W1^T, 4 tiles x 4 chained WMMAs ----
    v8f hv[4];
    #pragma unroll
    for (int tt = 0; tt < 4; ++tt) {
      v8f acc = zero8;
      #pragma unroll
      for (int c = 0; c < 4; ++c)
        acc = WMMA4(a1[c], w1v[tt][c], acc);
      hv[tt] = acc;
    }

    // ---- LIF updates (register elementwise) + spike stores to LDS ----
    #pragma unroll
    for (int tt = 0; tt < 4; ++tt) {
      v8f ss, si;
      #pragma unroll
      for (int r = 0; r < 8; ++r) {
        const float hval = hv[tt][r] + hb[tt];
        float ms = msyn[tt][r];
        const float rs = (ms - thr[tt] > 0.f) ? thr[tt] : 0.f;   // reset on PREV mem
        ms = beta[tt] * ms + hval - rs;
        msyn[tt][r] = ms;
        ss[r] = (ms - thr[tt] > 0.f) ? 1.f : 0.f;
        float mi = mint[tt][r];
        const float ri = (mi - thri[tt] > 0.f) ? thri[tt] : 0.f;
        mi = mi + hval - ri;                                      // beta = 1 integrator
        mint[tt][r] = mi;
        si[r] = (mi - thri[tt] > 0.f) ? 1.f : 0.f;
      }
      // GEMM1 D-layout: vgpr r = batch row r + 8*hi -> contiguous b128 stores
      float* ps = &spk[(16 * tt + n16) * 16 + (hi ? 8 : 0)];
      float* pi = ps + kH * 16;
      *(v4f*)(ps)     = __builtin_shufflevector(ss, ss, 0, 1, 2, 3);
      *(v4f*)(ps + 4) = __builtin_shufflevector(ss, ss, 4, 5, 6, 7);
      *(v4f*)(pi)     = __builtin_shufflevector(si, si, 0, 1, 2, 3);
      *(v4f*)(pi + 4) = __builtin_shufflevector(si, si, 4, 5, 6, 7);
    }
    __syncthreads();

    // ---- GEMM2: y^T[16ch x 16b] = W2_pad @ cat^T, K=128 as 32 chunks,
    //      4 accumulators to break the WMMA->WMMA RAW chain ----
    v8f acc0 = ybias, acc1 = zero8, acc2 = zero8, acc3 = zero8;
    #pragma unroll
    for (int c = 0; c < 32; ++c) {
      const int k = 4 * c + koff;
      v2f b; b.x = spk[k * 16 + n16]; b.y = spk[(k + 1) * 16 + n16];
      switch (c & 3) {
        case 0: acc0 = WMMA4(w2v[c], b, acc0); break;
        case 1: acc1 = WMMA4(w2v[c], b, acc1); break;
        case 2: acc2 = WMMA4(w2v[c], b, acc2); break;
        default: acc3 = WMMA4(w2v[c], b, acc3); break;
      }
    }
    __syncthreads();   // reads done before next iteration's stores
    yv = (acc0 + acc1) + (acc2 + acc3);

    // ---- store y[t]: lane n (low half) holds batch n channels 0..5 contiguously ----
    if (!hi) {
      float* po = out + ((size_t)(m0 + n16) * kT + t) * kOut;
      v2f s0; s0.x = yv[0]; s0.y = yv[1];
      v2f s1; s1.x = yv[2]; s1.y = yv[3];
      v2f s2; s2.x = yv[4]; s2.y = yv[5];
      *(v2f*)(po)     = s0;
      *(v2f*)(po + 2) = s1;
      *(v2f*)(po + 4) = s2;
    }
  }
}

extern "C" void kernel_launch(void* const* d_in, const int* in_sizes, int n_in,
                              void* d_out, int out_size, void* d_ws, size_t ws_size,
                              hipStream_t stream) {
  (void)n_in; (void)out_size; (void)d_ws; (void)ws_size;
  const float* x     = (const float*)d_in[0];
  const float* W1    = (const float*)d_in[1];
  const float* b1    = (const float*)d_in[2];
  const float* W2    = (const float*)d_in[3];
  const float* b2    = (const float*)d_in[4];
  const float* betas = (const float*)d_in[5];
  // d_in[6] = alphas: only multiplies a zero syn state in the reference -> unused.
  const float* thresholds = (const float*)d_in[7];
  const float* thr_int    = (const float*)d_in[8];

  const int B = in_sizes[0] / (kT * kDin);   // 2048
  snn_seq_kernel<<<dim3(B / 16), dim3(32), 0, stream>>>(
      x, W1, b1, W2, b2, betas, thresholds, thr_int, (float*)d_out);
}